// NURBS3D_66383014527124
// MI455X (gfx1250) — compile-verified
//
#include <hip/hip_runtime.h>

#define L 144            // eval points per axis
#define NCTRL 96         // control points per axis
#define DEG 3
#define KNOTS (NCTRL + DEG + 1)   // 100
#define CH 4             // live channels (weight channel is dead in the reference)

typedef float v2f __attribute__((ext_vector_type(2)));
typedef float v8f __attribute__((ext_vector_type(8)));

// ---------------------------------------------------------------------------
// Kernel 1: knots + sort(u) + find_span + Cox-de Boor basis for all 3 axes.
// Single block of 256 threads; all axes share one knot vector (n=96, deg=3).
// Writes dense basis matrices NuD/NvD [144 x 96] (4 nonzeros/row) and raw
// 4-tap weights + spans for the w axis (used by the gather stage).
// ---------------------------------------------------------------------------
__global__ void nurbs_basis_kernel(const float* __restrict__ u_in,
                                   const float* __restrict__ v_in,
                                   const float* __restrict__ w_in,
                                   float* __restrict__ NuD,
                                   float* __restrict__ NvD,
                                   float* __restrict__ Nw4,
                                   int* __restrict__ sw_out) {
    __shared__ float sU[KNOTS];
    __shared__ float s_u[L];
    const int tid = threadIdx.x;

    // Clamped-uniform knot vector, normalized cumsum closed form:
    // knot = [0]*4 ++ [1]*93 ++ [0]*3 ; cumsum[i] = clamp(i-3, 0, 93) ; /93
    if (tid < KNOTS) {
        int c = tid - DEG;
        c = c < 0 ? 0 : (c > (NCTRL - DEG) ? (NCTRL - DEG) : c);
        sU[tid] = (float)c / (float)(NCTRL - DEG);
    }
    // Zero the dense basis matrices (taps scattered below).
    for (int i = tid; i < L * NCTRL; i += 256) { NuD[i] = 0.0f; NvD[i] = 0.0f; }

    // Rank-sort u (144 elems, stable on ties).
    if (tid < L) {
        const float val = u_in[tid];
        int rank = 0;
        for (int j = 0; j < L; ++j) {
            const float o = u_in[j];
            rank += (o < val) || (o == val && j < tid);
        }
        s_u[rank] = val;
    }
    __syncthreads();

    // 3 axes x 144 points = 432 tasks.
    for (int task = tid; task < 3 * L; task += 256) {
        const int axis = task / L;
        const int p = task % L;
        const float v = (axis == 0) ? s_u[p] : (axis == 1 ? v_in[p] : w_in[p]);

        // find_span: argmin over masked (v - U[3+j]), j = 0..93 (first minimum).
        int best = 0;
        float bestv = 3.402823e+38f;
        for (int j = 0; j <= NCTRL - DEG; ++j) {
            const float diff = v - sU[DEG + j];
            const float m = (diff > 1e-8f) ? diff : 1.0f;
            if (m < bestv) { bestv = m; best = j; }
        }
        const int span = best + DEG;

        // Cox-de Boor recursion (with the reference's denom==0 -> 1e-4 quirk).
        float Ni[DEG + 1];
        Ni[0] = 1.0f; Ni[1] = 0.0f; Ni[2] = 0.0f; Ni[3] = 0.0f;
        for (int k = 1; k <= DEG; ++k) {
            float saved = 0.0f;
            for (int r = 0; r < k; ++r) {
                const float V1 = sU[span + r + 1];
                const float V2 = sU[span + 1 - k + r];
                const float denom = (V1 - v) + (v - V2);
                const float temp = (denom == 0.0f) ? 1e-4f : (Ni[r] / denom);
                Ni[r] = saved + (V1 - v) * temp;
                saved = (v - V2) * temp;
            }
            Ni[k] = saved;
        }

        if (axis == 0) {
            for (int i = 0; i <= DEG; ++i) {
                int col = span - DEG + i;
                col = col < 0 ? 0 : (col > NCTRL - 1 ? NCTRL - 1 : col);
                NuD[p * NCTRL + col] = Ni[i];
            }
        } else if (axis == 1) {
            for (int i = 0; i <= DEG; ++i) {
                int col = span - DEG + i;
                col = col < 0 ? 0 : (col > NCTRL - 1 ? NCTRL - 1 : col);
                NvD[p * NCTRL + col] = Ni[i];
            }
        } else {
            sw_out[p] = span;
            for (int i = 0; i <= DEG; ++i) Nw4[p * 4 + i] = Ni[i];
        }
    }
}

// ---------------------------------------------------------------------------
// Kernel 2: batched GEMM out[144, N] = W[144x96] * in[96, N] using
// V_WMMA_F32_16X16X4_F32. One wave per 16-wide N-tile; it walks all 9 M-tiles
// holding 9 v8f accumulators, so the streamed `in` operand is loaded once.
// Operand layouts per ISA 7.12.2:
//   A 16x4 f32 : lane half selects K-offset {0,2}; VGPR pair = K, K+1; row = lane&15
//   B 4x16 f32 : lane half selects K-offset {0,2}; VGPR pair = K, K+1; col = lane&15
//   C/D 16x16  : VGPR r holds rows {r, r+8} split across lane halves; col = lane&15
// ---------------------------------------------------------------------------
__global__ void nurbs_gemm_wmma(const float* __restrict__ W,
                                const float* __restrict__ in,
                                float* __restrict__ out,
                                int ncols, long inBatchStride, long outBatchStride,
                                int ntiles) {
    const int wave = threadIdx.x >> 5;
    const int lane = threadIdx.x & 31;
    const int ntile = blockIdx.x * (blockDim.x >> 5) + wave;
    if (ntile >= ntiles) return;

    const int n0 = ntile * 16;
    const int col = lane & 15;    // N-col within tile (also M-row for A operand)
    const int half = lane >> 4;   // K-offset selector {0 -> K0/K1, 1 -> K2/K3}
    const int kr = half * 2;

    const float* __restrict__ inb = in + (size_t)blockIdx.z * (size_t)inBatchStride;
    float* __restrict__ outb = out + (size_t)blockIdx.z * (size_t)outBatchStride;

    v8f acc[9] = {};

    for (int k0 = 0; k0 < NCTRL; k0 += 4) {
        // B operand: rows K = k0+kr, k0+kr+1 ; column n0+col (16-float coalesced rows)
        const float* bp = inb + (size_t)(k0 + kr) * (size_t)ncols + n0 + col;
        v2f b;
        b.x = bp[0];
        b.y = bp[ncols];
#pragma unroll
        for (int m = 0; m < 9; ++m) {
            // A operand: W row m*16+col, K = k0+kr, k0+kr+1
            const float* ap = W + (m * 16 + col) * NCTRL + k0 + kr;
            v2f a;
            a.x = ap[0];
            a.y = ap[1];
            acc[m] = __builtin_amdgcn_wmma_f32_16x16x4_f32(
                /*neg_a=*/false, a, /*neg_b=*/false, b,
                /*c_mod=*/(short)0, acc[m], /*reuse_a=*/false, /*reuse_b=*/false);
        }
    }

#pragma unroll
    for (int m = 0; m < 9; ++m) {
        const int mbase = m * 16 + half * 8;
#pragma unroll
        for (int r = 0; r < 8; ++r) {
            outb[(size_t)(mbase + r) * (size_t)ncols + n0 + col] = acc[m][r];
        }
    }
}

// ---------------------------------------------------------------------------
// Kernel 3: final w-axis contraction as a 4-tap float4 gather-MAC.
// out[u,v,ww,c] = sum_k Nw4[ww,k] * B[u,v,sw[ww]-3+k,c]   (memory-bound, 48 MB out)
// ---------------------------------------------------------------------------
__global__ void nurbs_stageC_kernel(const float* __restrict__ Bst,
                                    const float* __restrict__ Nw4,
                                    const int* __restrict__ sw,
                                    float* __restrict__ out) {
    const int idx = blockIdx.x * blockDim.x + threadIdx.x;
    const int total = L * L * L;
    if (idx >= total) return;

    const int ww = idx % L;
    const int uv = idx / L;           // u*L + v

    const float4* __restrict__ bp =
        reinterpret_cast<const float4*>(Bst + (size_t)uv * (NCTRL * CH));
    const int s = sw[ww] - DEG;

    float4 acc = make_float4(0.f, 0.f, 0.f, 0.f);
#pragma unroll
    for (int k = 0; k <= DEG; ++k) {
        int w = s + k;
        w = w < 0 ? 0 : (w > NCTRL - 1 ? NCTRL - 1 : w);
        const float nk = Nw4[ww * 4 + k];
        const float4 bv = bp[w];
        acc.x += nk * bv.x;
        acc.y += nk * bv.y;
        acc.z += nk * bv.z;
        acc.w += nk * bv.w;
    }
    reinterpret_cast<float4*>(out)[idx] = acc;
}

// ---------------------------------------------------------------------------
// d_in order: control_pts, weights, u_spline_space, v_spline_space,
//             w_spline_space, deg_x, deg_y, deg_z
// Weights are provably dead in the reference (channel 4 dropped, no division).
// ---------------------------------------------------------------------------
extern "C" void kernel_launch(void* const* d_in, const int* in_sizes, int n_in,
                              void* d_out, int out_size, void* d_ws, size_t ws_size,
                              hipStream_t stream) {
    const float* ctrl = (const float*)d_in[0];   // [96,96,96,4] row-major
    const float* u_in = (const float*)d_in[2];
    const float* v_in = (const float*)d_in[3];
    const float* w_in = (const float*)d_in[4];

    float* ws   = (float*)d_ws;
    float* NuD  = ws;                    // 144*96      = 13824
    float* NvD  = ws + 13824;            // 144*96      = 13824
    float* Nw4  = ws + 27648;            //  144*4      =   576
    int*   sw   = (int*)(ws + 28224);    //  144
    float* Abuf = ws + 28416;            // 144*96*96*4 = 5308416 (16B-aligned)
    float* Bbuf = Abuf + 5308416;        // 144*144*96*4 = 7962624 (16B-aligned)
    float* out  = (float*)d_out;         // 144^3*4 floats

    // 1) basis functions (1 block)
    nurbs_basis_kernel<<<1, 256, 0, stream>>>(u_in, v_in, w_in, NuD, NvD, Nw4, sw);

    // 2) stage A: A[u, (v,w,c)] = NuD[144x96] * ctrl[96 x 36864]
    {
        const int ncols = NCTRL * NCTRL * CH;        // 36864
        const int ntiles = ncols / 16;               // 2304
        dim3 grid((ntiles + 7) / 8, 1, 1);
        nurbs_gemm_wmma<<<grid, 256, 0, stream>>>(NuD, ctrl, Abuf, ncols, 0L, 0L, ntiles);
    }

    // 3) stage B (batched over u=144): B[u][vv,(w,c)] = NvD[144x96] * A[u][96 x 384]
    {
        const int ncols = NCTRL * CH;                // 384
        const int ntiles = ncols / 16;               // 24
        dim3 grid((ntiles + 7) / 8, 1, L);
        nurbs_gemm_wmma<<<grid, 256, 0, stream>>>(NvD, Abuf, Bbuf, ncols,
                                                  (long)(NCTRL * NCTRL * CH),  // 36864
                                                  (long)(L * NCTRL * CH),      // 55296
                                                  ntiles);
    }

    // 4) stage C: 4-tap gather producing the 48 MB output
    {
        const int total = L * L * L;
        nurbs_stageC_kernel<<<(total + 255) / 256, 256, 0, stream>>>(Bbuf, Nw4, sw, out);
    }
}